// LATTEConv_44547400794596
// MI455X (gfx1250) — compile-verified
//
#include <hip/hip_runtime.h>
#include <hip/hip_bf16.h>

typedef float v2f __attribute__((ext_vector_type(2)));
typedef float v8f __attribute__((ext_vector_type(8)));
typedef unsigned int u32x4 __attribute__((ext_vector_type(4)));
typedef int i32x8 __attribute__((ext_vector_type(8)));
typedef int i32x4 __attribute__((ext_vector_type(4)));

#define NEG_SLOPE 0.2f
#define LN_EPS 1e-5f

#if __has_builtin(__builtin_amdgcn_tensor_load_to_lds) && __has_builtin(__builtin_amdgcn_s_wait_tensorcnt)
#define USE_TDM 1
#else
#define USE_TDM 0
#endif

// ---------------------------------------------------------------------------
// K0: build combined B matrix [128 x 160] and combined bias [160]:
//   cols   0..127 : W_r                      (k = x@W_r + b_r)
//   cols 128..143 : vl[k,j] = sum_c W_l[k, h*32+c]*attn_l[j*32+c]  (j=r*4+h, 12 used)
//   cols 144..159 : vr[k,j] = sum_c W_r[k, h*32+c]*attn_r[j*32+c]
// biasc: b_r | (b_l . attn_l[j]) | (b_r . attn_r[j])
// ---------------------------------------------------------------------------
__global__ void k0_build(const float* __restrict__ Wl, const float* __restrict__ bl,
                         const float* __restrict__ Wr, const float* __restrict__ br,
                         const float* __restrict__ al, const float* __restrict__ ar,
                         float* __restrict__ Bc, float* __restrict__ biasc) {
    int t = threadIdx.x;  // 256 threads, 1 block
    for (int idx = t; idx < 128 * 160; idx += 256) {
        int k = idx / 160, c = idx % 160;
        float v = 0.f;
        if (c < 128) {
            v = Wr[k * 128 + c];
        } else if (c < 144) {
            int j = c - 128;
            if (j < 12) {
                float s = 0.f;
                #pragma unroll 8
                for (int cc = 0; cc < 32; ++cc)
                    s += Wl[k * 128 + (j & 3) * 32 + cc] * al[j * 32 + cc];
                v = s;
            }
        } else {
            int j = c - 144;
            if (j < 12) {
                float s = 0.f;
                #pragma unroll 8
                for (int cc = 0; cc < 32; ++cc)
                    s += Wr[k * 128 + (j & 3) * 32 + cc] * ar[j * 32 + cc];
                v = s;
            }
        }
        Bc[idx] = v;
    }
    if (t < 160) {
        float v = 0.f;
        if (t < 128) {
            v = br[t];
        } else if (t < 144) {
            int j = t - 128;
            if (j < 12) { for (int cc = 0; cc < 32; ++cc) v += bl[(j & 3) * 32 + cc] * al[j * 32 + cc]; }
        } else {
            int j = t - 144;
            if (j < 12) { for (int cc = 0; cc < 32; ++cc) v += br[(j & 3) * 32 + cc] * ar[j * 32 + cc]; }
        }
        biasc[t] = v;
    }
}

// ---------------------------------------------------------------------------
// TDM helper: DMA a [rows x 128] fp32 tile (row-major, stride 128) from global
// into LDS at byte offset lds_off. D# per CDNA5 ISA ch.8 (2D tensor, groups
// 2/3 zero). A wave passing count=0 issues a NULL-tensor NOP, which lets all
// waves execute the instruction in uniform control flow while only wave 0
// actually transfers data (TDM ignores EXEC).
// ---------------------------------------------------------------------------
#if USE_TDM
__device__ __forceinline__ void tdm_load_strip(const float* gsrc, unsigned lds_byte_off,
                                               int valid_rows, int count) {
    unsigned long long ga = (unsigned long long)(uintptr_t)gsrc;
    u32x4 g0;
    g0[0] = (unsigned)count;                                    // count (0 = NULL tensor)
    g0[1] = lds_byte_off;                                       // lds_addr
    g0[2] = (unsigned)(ga & 0xffffffffull);                     // global_addr[31:0]
    g0[3] = (unsigned)((ga >> 32) & 0x1ffffffull) | 0x80000000u; // addr[56:32] | type=2
    i32x8 g1;
    g1[0] = (2 << 16);                  // data_size = 4B; mask/flags = 0
    g1[1] = (128 << 16);                // tensor_dim0[15:0] = 128
    g1[2] = (valid_rows << 16);         // tensor_dim0[31:16]=0 | tensor_dim1[15:0]
    g1[3] = (128 << 16);                // tensor_dim1[31:16]=0 | tile_dim0 = 128
    g1[4] = 16;                         // tile_dim1 = 16 (OOB rows read as zero), tile_dim2 = 0
    g1[5] = 128;                        // tensor_dim0_stride[31:0] = 128 elements
    g1[6] = 0;                          // stride0[47:32] | stride1[15:0]
    g1[7] = 0;                          // stride1[47:16]
    i32x4 z4 = {0, 0, 0, 0};
    i32x8 z8 = {0, 0, 0, 0, 0, 0, 0, 0};
    // 6-arg form (upstream clang-23 / therock-10.0 toolchain)
    __builtin_amdgcn_tensor_load_to_lds(g0, g1, z4, z4, z8, 0);
}
__device__ __forceinline__ unsigned lds_byte_offset(const void* p) {
    // low 32 bits of a generic LDS pointer are the LDS byte address
    return (unsigned)(unsigned long long)(uintptr_t)p;
}
#endif

// ---------------------------------------------------------------------------
// K1: fused WMMA GEMM.  [N x 128] @ [128 x 160] + bias.
// Block = 320 threads = 10 waves; wave w owns output col-tile w (16 cols).
// Each block processes 5 strips of 16 rows; x strip staged in LDS via TDM
// (double-buffered, TENSORcnt-pipelined), per-wave B strip (128x16 fp32) held
// in 64 VGPRs. Uses V_WMMA_F32_16X16X4_F32, 32 per tile (K=128).
// ---------------------------------------------------------------------------
#define K1_WAVES 10
#define K1_STRIPS 5
__global__ __launch_bounds__(320) void k1_gemm(
        const float* __restrict__ x, const float* __restrict__ Bc,
        const float* __restrict__ biasc, float* __restrict__ kmat,
        float* __restrict__ ql, float* __restrict__ kr, int N) {
    __shared__ __align__(16) float xs[2][16 * 128];
    const int tid  = threadIdx.x;
    const int wave = tid >> 5;
    const int lane = tid & 31;
    const int lo   = lane & 15;
    const int hi   = lane >> 4;
    const int col  = wave * 16 + lo;

    // Preload this wave's B strip into registers:
    // b0(j) = Bc[4j + 2*hi][col], b1(j) = Bc[4j + 2*hi + 1][col]
    float breg[64];
    #pragma unroll
    for (int j = 0; j < 32; ++j) {
        int k0 = 4 * j + 2 * hi;
        breg[2 * j]     = Bc[(size_t)k0 * 160 + col];
        breg[2 * j + 1] = Bc[(size_t)(k0 + 1) * 160 + col];
    }
    const float cb = biasc[col];

    const long base_row = (long)blockIdx.x * (16 * K1_STRIPS);

#if USE_TDM
    const int cnt0 = (wave == 0) ? 1 : 0;   // only wave 0 moves data
    {
        long row0 = base_row;
        int vr = (int)((N - row0) < 16 ? (N - row0) : 16);
        tdm_load_strip(x + row0 * 128, lds_byte_offset(&xs[0][0]), vr, cnt0);
    }
#endif

    for (int s = 0; s < K1_STRIPS; ++s) {
        const long row0 = base_row + s * 16;
#if USE_TDM
        const float* xbuf = xs[s & 1];
        if (s + 1 < K1_STRIPS) {
            long row1 = row0 + 16;
            int vr = (int)((N - row1) < 16 ? (N - row1) : 16);
            tdm_load_strip(x + row1 * 128, lds_byte_offset(&xs[(s + 1) & 1][0]), vr, cnt0);
            __builtin_amdgcn_s_wait_tensorcnt(1);   // strip s complete, s+1 in flight
        } else {
            __builtin_amdgcn_s_wait_tensorcnt(0);
        }
        __syncthreads();
#else
        const float* xbuf = xs[0];
        // fallback: cooperative load of 16x128 fp32 strip (512 float4)
        for (int idx = tid; idx < 16 * 32; idx += 320) {
            int  rr  = idx >> 5, cc4 = idx & 31;
            long r   = row0 + rr;
            if (r >= N) r = N - 1;
            ((float4*)xs[0])[idx] = ((const float4*)(x + r * 128))[cc4];
        }
        __syncthreads();
#endif

        v8f c = {cb, cb, cb, cb, cb, cb, cb, cb};
        #pragma unroll
        for (int j = 0; j < 32; ++j) {
            // A frag: row = lo, K = 4j + 2*hi (+1)  -> adjacent pair, ds_load_b64
            v2f a = *(const v2f*)(&xbuf[lo * 128 + 4 * j + 2 * hi]);
            v2f b = {breg[2 * j], breg[2 * j + 1]};
            c = __builtin_amdgcn_wmma_f32_16x16x4_f32(
                    /*neg_a=*/false, a, /*neg_b=*/false, b,
                    /*c_mod=*/(short)0, c, /*reuse_a=*/false, /*reuse_b=*/false);
        }

        // D layout: VGPR g -> row = row0 + g + 8*hi, col = wave*16 + lo
        #pragma unroll
        for (int g = 0; g < 8; ++g) {
            long row = row0 + g + 8 * hi;
            if (row < N) {
                float v = c[g];
                if (wave < 8)       kmat[row * 128 + col] = v;
                else if (wave == 8) { if (lo < 12) ql[row * 12 + lo] = v; }
                else                { if (lo < 12) kr[row * 12 + lo] = v; }
            }
        }
        __syncthreads();
    }
}

// ---------------------------------------------------------------------------
// Edge passes. att[e,h] = leakyrelu(ql[src,r,h] + kr[dst,r,h])
// ---------------------------------------------------------------------------
__device__ __forceinline__ unsigned f2ord(float f) {
    unsigned b = __float_as_uint(f);
    return (b & 0x80000000u) ? ~b : (b | 0x80000000u);
}
__device__ __forceinline__ float ord2f(unsigned o) {
    return (o & 0x80000000u) ? __uint_as_float(o & 0x7fffffffu) : __uint_as_float(~o);
}
__device__ __forceinline__ float4 edge_att(const float4* ql4, const float4* kr4,
                                           int s, int d, int r) {
    float4 a = ql4[(size_t)s * 3 + r];
    float4 b = kr4[(size_t)d * 3 + r];
    float4 t;
    t.x = a.x + b.x; t.y = a.y + b.y; t.z = a.z + b.z; t.w = a.w + b.w;
    t.x = t.x > 0.f ? t.x : NEG_SLOPE * t.x;
    t.y = t.y > 0.f ? t.y : NEG_SLOPE * t.y;
    t.z = t.z > 0.f ? t.z : NEG_SLOPE * t.z;
    t.w = t.w > 0.f ? t.w : NEG_SLOPE * t.w;
    return t;
}

__global__ __launch_bounds__(256) void k2_amax(
        const int* __restrict__ esrc, const int* __restrict__ edst,
        const float4* __restrict__ ql4, const float4* __restrict__ kr4,
        unsigned* __restrict__ amax, int N, int E) {
    int i = blockIdx.x * 256 + threadIdx.x;
    int r = blockIdx.y;
    if (i >= E) return;
    int s = esrc[(size_t)r * E + i], d = edst[(size_t)r * E + i];
    float4 t = edge_att(ql4, kr4, s, d, r);
    unsigned* base = amax + ((size_t)r * N + d) * 4;
    atomicMax(base + 0, f2ord(t.x));
    atomicMax(base + 1, f2ord(t.y));
    atomicMax(base + 2, f2ord(t.z));
    atomicMax(base + 3, f2ord(t.w));
}

__global__ __launch_bounds__(256) void k3_denom(
        const int* __restrict__ esrc, const int* __restrict__ edst,
        const float4* __restrict__ ql4, const float4* __restrict__ kr4,
        const unsigned* __restrict__ amax, float* __restrict__ denom, int N, int E) {
    int i = blockIdx.x * 256 + threadIdx.x;
    int r = blockIdx.y;
    if (i >= E) return;
    int s = esrc[(size_t)r * E + i], d = edst[(size_t)r * E + i];
    float4 t = edge_att(ql4, kr4, s, d, r);
    size_t base = ((size_t)r * N + d) * 4;
    const unsigned* am = amax + base;
    float* dn = denom + base;
    float att[4] = {t.x, t.y, t.z, t.w};
    #pragma unroll
    for (int h = 0; h < 4; ++h) {
        unsigned raw = am[h];
        float m = raw ? ord2f(raw) : 0.0f;  // reference: non-finite amax -> 0
        atomicAdd(dn + h, __expf(att[h] - m));
    }
}

__global__ __launch_bounds__(256) void k4_salpha(
        const int* __restrict__ esrc, const int* __restrict__ edst,
        const float4* __restrict__ ql4, const float4* __restrict__ kr4,
        const unsigned* __restrict__ amax, const float* __restrict__ denom,
        float* __restrict__ salpha, int N, int E) {
    int i = blockIdx.x * 256 + threadIdx.x;
    int r = blockIdx.y;
    if (i >= E) return;
    int s = esrc[(size_t)r * E + i], d = edst[(size_t)r * E + i];
    float4 t = edge_att(ql4, kr4, s, d, r);
    size_t base = ((size_t)r * N + d) * 4;
    const unsigned* am = amax + base;
    const float* dn = denom + base;
    float* sa = salpha + base;
    float att[4] = {t.x, t.y, t.z, t.w};
    #pragma unroll
    for (int h = 0; h < 4; ++h) {
        unsigned raw = am[h];
        float m  = raw ? ord2f(raw) : 0.0f;
        float ex = __expf(att[h] - m);
        atomicAdd(sa + h, ex / fmaxf(dn[h], 1e-16f));
    }
}

// ---------------------------------------------------------------------------
// K5: per-node combine + beta relation-attention + layernorm.
// One 128-thread block per node; wave32 => wave w == head h. Message for
// relation r is k[n]*salpha[r,n,h] (segment_sum(alpha*k[dst]) with k[dst]
// constant per segment).
// ---------------------------------------------------------------------------
__device__ __forceinline__ float wred(float v) {
    #pragma unroll
    for (int off = 16; off > 0; off >>= 1) v += __shfl_xor(v, off, 32);
    return v;
}

__global__ __launch_bounds__(128) void k5_combine(
        const float* __restrict__ kmat, const float* __restrict__ salpha,
        const float* __restrict__ rl, const float* __restrict__ rr,
        const float* __restrict__ gamma, const float* __restrict__ betaln,
        float* __restrict__ out, int N) {
    const int n = blockIdx.x;
    const int d = threadIdx.x;    // 0..127 over (h, c)
    const int h = d >> 5;
    float kv = kmat[(size_t)n * 128 + d];

    float hv[4];
    hv[3] = kv;
    #pragma unroll
    for (int r = 0; r < 3; ++r)
        hv[r] = kv * salpha[((size_t)r * N + n) * 4 + h];

    float rlv = rl[d], rrv = rr[d];
    float A = wred(fmaxf(kv * rlv, 0.f));   // sum_c relu(query*rel_attn_l) per head
    float sc[4];
    #pragma unroll
    for (int j = 0; j < 4; ++j)
        sc[j] = A + wred(fmaxf(hv[j] * rrv, 0.f));

    float m = fmaxf(fmaxf(sc[0], sc[1]), fmaxf(sc[2], sc[3]));
    float e[4], ssum = 0.f;
    #pragma unroll
    for (int j = 0; j < 4; ++j) { e[j] = __expf(sc[j] - m); ssum += e[j]; }
    float inv = 1.f / ssum;
    float o = 0.f;
    #pragma unroll
    for (int j = 0; j < 4; ++j) o += hv[j] * (e[j] * inv);

    // layernorm across 128 dims (4 waves)
    __shared__ float red[8];
    float s1 = wred(o), s2 = wred(o * o);
    if ((d & 31) == 0) { red[h] = s1; red[4 + h] = s2; }
    __syncthreads();
    float S1 = red[0] + red[1] + red[2] + red[3];
    float S2 = red[4] + red[5] + red[6] + red[7];
    float mu  = S1 * (1.f / 128.f);
    float var = S2 * (1.f / 128.f) - mu * mu;
    out[(size_t)n * 128 + d] = (o - mu) * rsqrtf(var + LN_EPS) * gamma[d] + betaln[d];
}

// ---------------------------------------------------------------------------
extern "C" void kernel_launch(void* const* d_in, const int* in_sizes, int n_in,
                              void* d_out, int out_size, void* d_ws, size_t ws_size,
                              hipStream_t stream) {
    const float* x    = (const float*)d_in[0];
    const int*   esrc = (const int*)  d_in[1];
    const int*   edst = (const int*)  d_in[2];
    const float* Wl   = (const float*)d_in[3];
    const float* bl   = (const float*)d_in[4];
    const float* Wr   = (const float*)d_in[5];
    const float* br   = (const float*)d_in[6];
    const float* al   = (const float*)d_in[7];
    const float* ar   = (const float*)d_in[8];
    const float* rl   = (const float*)d_in[9];
    const float* rr   = (const float*)d_in[10];
    const float* lng  = (const float*)d_in[11];
    const float* lnb  = (const float*)d_in[12];

    const int N = in_sizes[0] / 128;   // 100000
    const int E = in_sizes[1] / 3;     // 400000

    // workspace carve-up (256B aligned); total ~80 MB
    char* ws = (char*)d_ws;
    size_t off = 0;
    auto carve = [&](size_t bytes) -> char* {
        char* p = ws + off;
        off = (off + bytes + 255) & ~(size_t)255;
        return p;
    };
    float*    Bc     = (float*)   carve(128 * 160 * sizeof(float));
    float*    biasc  = (float*)   carve(160 * sizeof(float));
    float*    ql     = (float*)   carve((size_t)N * 12 * sizeof(float));
    float*    kr     = (float*)   carve((size_t)N * 12 * sizeof(float));
    float*    kmat   = (float*)   carve((size_t)N * 128 * sizeof(float));
    unsigned* amax   = (unsigned*)carve((size_t)3 * N * 4 * sizeof(unsigned));
    float*    denom  = (float*)   carve((size_t)3 * N * 4 * sizeof(float));
    float*    salpha = (float*)   carve((size_t)3 * N * 4 * sizeof(float));

    // zero accumulators every call (harness does not re-poison between replays)
    (void)hipMemsetAsync(amax,   0, (size_t)3 * N * 4 * sizeof(unsigned), stream);
    (void)hipMemsetAsync(denom,  0, (size_t)3 * N * 4 * sizeof(float),    stream);
    (void)hipMemsetAsync(salpha, 0, (size_t)3 * N * 4 * sizeof(float),    stream);

    k0_build<<<1, 256, 0, stream>>>(Wl, bl, Wr, br, al, ar, Bc, biasc);

    int blocks1 = (N + 16 * K1_STRIPS - 1) / (16 * K1_STRIPS);
    k1_gemm<<<blocks1, 32 * K1_WAVES, 0, stream>>>(x, Bc, biasc, kmat, ql, kr, N);

    dim3 ge((E + 255) / 256, 3);
    const float4* ql4 = (const float4*)ql;
    const float4* kr4 = (const float4*)kr;
    k2_amax  <<<ge, 256, 0, stream>>>(esrc, edst, ql4, kr4, amax, N, E);
    k3_denom <<<ge, 256, 0, stream>>>(esrc, edst, ql4, kr4, amax, denom, N, E);
    k4_salpha<<<ge, 256, 0, stream>>>(esrc, edst, ql4, kr4, amax, denom, salpha, N, E);

    k5_combine<<<N, 128, 0, stream>>>(kmat, salpha, rl, rr, lng, lnb, (float*)d_out, N);
}